// BotGCN_5531917877303
// MI455X (gfx1250) — compile-verified
//
#include <hip/hip_runtime.h>

#define NN 100000
#define EE 3200000
#define DD 128
#define SLOPE 0.01f

typedef __attribute__((ext_vector_type(2))) float v2f;
typedef __attribute__((ext_vector_type(8))) float v8f;

__device__ __forceinline__ float lrelu(float v) { return v > 0.0f ? v : SLOPE * v; }

// D = A(16x4,f32) * B(4x16,f32) + C(16x16,f32)   -> v_wmma_f32_16x16x4_f32
__device__ __forceinline__ v8f wmma4(v2f a, v2f b, v8f c) {
    return __builtin_amdgcn_wmma_f32_16x16x4_f32(false, a, false, b, (short)0, c, false, false);
}

// ---------------------------------------------------------------------------
// Feature encoder: h[:,0:32]=lrelu(des@Wd+bd), h[:,32:64]=lrelu(tweet@Wt+bt),
//                  h[:,64:96]=lrelu(num@Wn+bn), h[:,96:128]=lrelu(cat@Wc+bc)
// 32 rows/block, 8 waves: wave -> (matrix = w>>2, row tile = (w>>1)&1, col tile = w&1).
// K in 64-wide chunks; A and W both staged in LDS; W transposed ([col][k]) so the
// per-k-step B fragment is a contiguous b64, stride 68 -> lane banks 4m (conflict-free).
// ---------------------------------------------------------------------------
__global__ __launch_bounds__(256) void encode_kernel(
    const float* __restrict__ des, const float* __restrict__ tweet,
    const float* __restrict__ nump, const float* __restrict__ catp,
    const float* __restrict__ Wd, const float* __restrict__ bd,
    const float* __restrict__ Wt, const float* __restrict__ bt,
    const float* __restrict__ Wn, const float* __restrict__ bn,
    const float* __restrict__ Wc, const float* __restrict__ bc,
    float* __restrict__ h)
{
    __shared__ float sA[2][32 * 68];     // activations chunk: 32 rows x 64 K
    __shared__ float sBt[2][32 * 68];    // weights chunk, transposed: 32 cols x 64 K
    const int row0 = blockIdx.x * 32;
    const int tid  = threadIdx.x;
    const int wave = tid >> 5;
    const int lane = tid & 31;
    const int m    = lane & 15;
    const int koff = (lane >> 4) << 1;   // 0 or 2 (A/B K sub-offset)
    const int hi8  = (lane >> 4) << 3;   // 0 or 8 (D row offset)

    const int mt = wave >> 2;            // 0: des, 1: tweet
    const int rt = (wave >> 1) & 1;      // row tile within block
    const int ct = wave & 1;             // col tile within 32-wide output

    // loader roles: threads 0-127 stage des+Wd, threads 128-255 stage tweet+Wt
    const int    lm   = tid >> 7;
    const int    lt   = tid & 127;
    const float* lsrc = lm ? tweet : des;
    const float* lW   = lm ? Wt : Wd;
    float* sAl  = sA[lm];
    float* sBtl = sBt[lm];

    v8f acc = {};
    for (int kc = 0; kc < 12; ++kc) {    // K = 12 * 64 = 768
        const int k0 = kc << 6;
        __syncthreads();
        // stage A: 32 rows x 64 floats = 2048 floats per matrix (128 thr x 4 float4)
        for (int i = 0; i < 4; ++i) {
            int idx = ((i << 7) + lt) << 2;            // 0..2044
            int r = idx >> 6, c = idx & 63;
            const float4 v = *(const float4*)(lsrc + (size_t)(row0 + r) * 768 + k0 + c);
            *(float4*)(sAl + r * 68 + c) = v;
        }
        // stage W chunk transposed: 64 K x 32 cols = 2048 floats per matrix
        for (int i = 0; i < 4; ++i) {
            int idx = ((i << 7) + lt) << 2;
            int k = idx >> 5, c = idx & 31;
            const float4 v = *(const float4*)(lW + (size_t)(k0 + k) * 32 + c);
            sBtl[(c + 0) * 68 + k] = v.x;
            sBtl[(c + 1) * 68 + k] = v.y;
            sBtl[(c + 2) * 68 + k] = v.z;
            sBtl[(c + 3) * 68 + k] = v.w;
        }
        if (kc + 1 < 12) {   // hint the next streaming chunk (global_prefetch_b8)
            __builtin_prefetch(lsrc + (size_t)(row0 + (lt & 31)) * 768 + k0 + 64, 0, 0);
        }
        __syncthreads();
        const float* sAr = &sA[mt][(rt * 16 + m) * 68];
        const float* sBr = &sBt[mt][(ct * 16 + m) * 68];
        #pragma unroll
        for (int k = 0; k < 16; ++k) {
            const int kb = (k << 2) + koff;
            v2f a = *(const v2f*)(sAr + kb);
            v2f b = *(const v2f*)(sBr + kb);
            acc = wmma4(a, b, acc);
        }
    }
    const float* bb = mt ? bt : bd;
    const int wcol = (ct << 4) + m;
    const float bv = bb[wcol];
    const int hcol = (mt << 5) + wcol;
    #pragma unroll
    for (int r = 0; r < 8; ++r) {
        int row = row0 + rt * 16 + r + hi8;
        h[(size_t)row * DD + hcol] = lrelu(acc[r] + bv);
    }
    // num (K=6) and cat (K=11) scalar tail: 2048 outputs / 256 threads
    for (int j = 0; j < 8; ++j) {
        int idx = tid + (j << 8);
        int c = idx & 31, r = (idx >> 5) & 31;
        int row = row0 + r;
        if (idx < 1024) {
            float a = bn[c];
            #pragma unroll
            for (int k = 0; k < 6; ++k)  a += nump[(size_t)row * 6 + k]  * Wn[k * 32 + c];
            h[(size_t)row * DD + 64 + c] = lrelu(a);
        } else {
            float a = bc[c];
            #pragma unroll
            for (int k = 0; k < 11; ++k) a += catp[(size_t)row * 11 + k] * Wc[k * 32 + c];
            h[(size_t)row * DD + 96 + c] = lrelu(a);
        }
    }
}

// ---------------------------------------------------------------------------
// out[n,128] = act(X[n,128] @ W[128,128] + bias)  via f32 WMMA, 64 rows/block.
// 8 waves: wave -> (row tile = wave>>1, column half = wave&1), 4 col tiles each.
// Both A and W staged in LDS (W transposed), K chunked by 64 (52 KB LDS total).
// ---------------------------------------------------------------------------
__global__ __launch_bounds__(256) void gemm128_kernel(
    const float* __restrict__ X, const float* __restrict__ W,
    const float* __restrict__ bias, int act, int n_rows,
    float* __restrict__ out)
{
    __shared__ float sA[64 * 68];        // A chunk: 64 rows x 64 K
    __shared__ float sBt[128 * 68];      // W chunk transposed: 128 cols x 64 K
    const int row0 = blockIdx.x * 64;
    const int tid  = threadIdx.x;
    const int wave = tid >> 5;
    const int lane = tid & 31;
    const int m    = lane & 15;
    const int koff = (lane >> 4) << 1;
    const int hi8  = (lane >> 4) << 3;
    const int rt   = wave >> 1;          // row tile 0..3
    const int ch   = (wave & 1) << 6;    // col half: 0 or 64

    v8f acc[4] = {{}, {}, {}, {}};
    for (int kc = 0; kc < 2; ++kc) {
        const int k0 = kc << 6;
        __syncthreads();
        // stage A chunk: 64 rows x 64 floats = 4096 floats (256 thr x 4 float4)
        for (int i = 0; i < 4; ++i) {
            int idx = ((i << 8) + tid) << 2;           // 0..4092
            int r = idx >> 6, c = idx & 63;
            int row = row0 + r;
            float4 v = {0.f, 0.f, 0.f, 0.f};
            if (row < n_rows) v = *(const float4*)(X + (size_t)row * DD + k0 + c);
            *(float4*)(sA + r * 68 + c) = v;
        }
        // stage W chunk transposed: 64 K x 128 cols = 8192 floats (256 thr x 8 float4)
        for (int i = 0; i < 8; ++i) {
            int idx = ((i << 8) + tid) << 2;           // 0..8188
            int k = idx >> 7, c = idx & 127;
            const float4 v = *(const float4*)(W + (size_t)(k0 + k) * DD + c);
            sBt[(c + 0) * 68 + k] = v.x;
            sBt[(c + 1) * 68 + k] = v.y;
            sBt[(c + 2) * 68 + k] = v.z;
            sBt[(c + 3) * 68 + k] = v.w;
        }
        __syncthreads();
        const float* sAr = sA + (rt * 16 + m) * 68;
        #pragma unroll
        for (int k = 0; k < 16; ++k) {
            const int kb = (k << 2) + koff;
            v2f a = *(const v2f*)(sAr + kb);
            #pragma unroll
            for (int ct = 0; ct < 4; ++ct) {
                v2f b = *(const v2f*)(sBt + (ch + ct * 16 + m) * 68 + kb);
                acc[ct] = wmma4(a, b, acc[ct]);
            }
        }
    }
    #pragma unroll
    for (int ct = 0; ct < 4; ++ct) {
        int col = ch + ct * 16 + m;
        float bv = bias ? bias[col] : 0.0f;
        #pragma unroll
        for (int r = 0; r < 8; ++r) {
            int row = row0 + rt * 16 + r + hi8;
            if (row < n_rows) {
                float v = acc[ct][r] + bv;
                if (act) v = lrelu(v);
                out[(size_t)row * DD + col] = v;
            }
        }
    }
}

// ---------------------------------------------------------------------------
// GCN aggregation helpers
// ---------------------------------------------------------------------------
__global__ void deg_init_kernel(float* deg, int n) {
    int i = blockIdx.x * blockDim.x + threadIdx.x;
    if (i < n) deg[i] = 1.0f;                          // self-loop
}
__global__ void deg_edges_kernel(const int* __restrict__ dst, float* deg, int e) {
    int i = blockIdx.x * blockDim.x + threadIdx.x;
    if (i < e) atomicAdd(&deg[dst[i]], 1.0f);
}
__global__ void dinv_kernel(const float* __restrict__ deg, float* dinv, int n) {
    int i = blockIdx.x * blockDim.x + threadIdx.x;
    if (i < n) { float d = deg[i]; dinv[i] = d > 0.0f ? rsqrtf(d) : 0.0f; }
}
// out[i,c] = dinv[i]^2 * xw[i,c] + bias[c]   (self-loop term + bias)
__global__ void selfloop_bias_kernel(const float* __restrict__ xw,
                                     const float* __restrict__ dinv,
                                     const float* __restrict__ bias,
                                     float* __restrict__ out, int n) {
    int idx = blockIdx.x * blockDim.x + threadIdx.x;
    if (idx < n * DD) {
        int i = idx >> 7, c = idx & 127;
        float di = dinv[i];
        out[idx] = di * di * xw[idx] + bias[c];
    }
}
// wave per edge: out[dst] += dinv[s]*dinv[d] * xw[src]  (128 f32 via 32 lanes x float4)
__global__ __launch_bounds__(256) void scatter_kernel(
    const float* __restrict__ xw, const float* __restrict__ dinv,
    const int* __restrict__ src, const int* __restrict__ dst,
    float* __restrict__ out, int e)
{
    int eid = blockIdx.x * 8 + (threadIdx.x >> 5);
    if (eid >= e) return;
    int s = src[eid], d = dst[eid];
    float nrm = dinv[s] * dinv[d];
    int lane = threadIdx.x & 31;
    const float4 v = *(const float4*)(xw + (size_t)s * DD + lane * 4);
    float* o = out + (size_t)d * DD + lane * 4;
    atomicAdd(o + 0, v.x * nrm);
    atomicAdd(o + 1, v.y * nrm);
    atomicAdd(o + 2, v.z * nrm);
    atomicAdd(o + 3, v.w * nrm);
}
// wave per node: out[n,2] = t[n,:] @ W2[128,2] + b2
__global__ __launch_bounds__(256) void head_kernel(
    const float* __restrict__ t, const float* __restrict__ W2,
    const float* __restrict__ b2, float* __restrict__ out, int n)
{
    int node = blockIdx.x * 8 + (threadIdx.x >> 5);
    if (node >= n) return;
    int lane = threadIdx.x & 31;
    const float4 v = *(const float4*)(t + (size_t)node * DD + lane * 4);
    int k = lane * 4;
    float a0 = v.x * W2[(k + 0) * 2]     + v.y * W2[(k + 1) * 2]
             + v.z * W2[(k + 2) * 2]     + v.w * W2[(k + 3) * 2];
    float a1 = v.x * W2[(k + 0) * 2 + 1] + v.y * W2[(k + 1) * 2 + 1]
             + v.z * W2[(k + 2) * 2 + 1] + v.w * W2[(k + 3) * 2 + 1];
    for (int off = 16; off > 0; off >>= 1) {
        a0 += __shfl_xor(a0, off, 32);
        a1 += __shfl_xor(a1, off, 32);
    }
    if (lane == 0) {
        out[(size_t)node * 2 + 0] = a0 + b2[0];
        out[(size_t)node * 2 + 1] = a1 + b2[1];
    }
}

// ---------------------------------------------------------------------------
extern "C" void kernel_launch(void* const* d_in, const int* in_sizes, int n_in,
                              void* d_out, int out_size, void* d_ws, size_t ws_size,
                              hipStream_t stream) {
    const float* des   = (const float*)d_in[0];
    const float* tweet = (const float*)d_in[1];
    const float* nump  = (const float*)d_in[2];
    const float* catp  = (const float*)d_in[3];
    const int*   eidx  = (const int*)d_in[4];
    // d_in[5] edge_type: unused by the reference math
    const float* Wd  = (const float*)d_in[6];  const float* bd  = (const float*)d_in[7];
    const float* Wt  = (const float*)d_in[8];  const float* bt  = (const float*)d_in[9];
    const float* Wn  = (const float*)d_in[10]; const float* bn  = (const float*)d_in[11];
    const float* Wc  = (const float*)d_in[12]; const float* bc  = (const float*)d_in[13];
    const float* Win = (const float*)d_in[14]; const float* bin = (const float*)d_in[15];
    const float* Wg1 = (const float*)d_in[16]; const float* bg1 = (const float*)d_in[17];
    const float* Wg2 = (const float*)d_in[18]; const float* bg2 = (const float*)d_in[19];
    const float* Wo1 = (const float*)d_in[20]; const float* bo1 = (const float*)d_in[21];
    const float* Wo2 = (const float*)d_in[22]; const float* bo2 = (const float*)d_in[23];

    const int n = NN, e = EE;
    const int* src = eidx;
    const int* dst = eidx + e;

    float* buf0 = (float*)d_ws;
    float* buf1 = buf0 + (size_t)n * DD;
    float* buf2 = buf1 + (size_t)n * DD;
    float* deg  = buf2 + (size_t)n * DD;
    float* dinv = deg + n;

    // feature encoder
    encode_kernel<<<n / 32, 256, 0, stream>>>(des, tweet, nump, catp,
                                              Wd, bd, Wt, bt, Wn, bn, Wc, bc, buf1);
    gemm128_kernel<<<(n + 63) / 64, 256, 0, stream>>>(buf1, Win, bin, 1, n, buf0);
    // degree normalization (shared by both convs)
    deg_init_kernel<<<(n + 255) / 256, 256, 0, stream>>>(deg, n);
    deg_edges_kernel<<<(e + 255) / 256, 256, 0, stream>>>(dst, deg, e);
    dinv_kernel<<<(n + 255) / 256, 256, 0, stream>>>(deg, dinv, n);
    // GCNConv 1
    gemm128_kernel<<<(n + 63) / 64, 256, 0, stream>>>(buf0, Wg1, nullptr, 0, n, buf1);
    selfloop_bias_kernel<<<((n * DD) + 255) / 256, 256, 0, stream>>>(buf1, dinv, bg1, buf2, n);
    scatter_kernel<<<(e + 7) / 8, 256, 0, stream>>>(buf1, dinv, src, dst, buf2, e);
    // GCNConv 2
    gemm128_kernel<<<(n + 63) / 64, 256, 0, stream>>>(buf2, Wg2, nullptr, 0, n, buf1);
    selfloop_bias_kernel<<<((n * DD) + 255) / 256, 256, 0, stream>>>(buf1, dinv, bg2, buf0, n);
    scatter_kernel<<<(e + 7) / 8, 256, 0, stream>>>(buf1, dinv, src, dst, buf0, e);
    // output head
    gemm128_kernel<<<(n + 63) / 64, 256, 0, stream>>>(buf0, Wo1, bo1, 1, n, buf1);
    head_kernel<<<(n + 7) / 8, 256, 0, stream>>>(buf1, Wo2, bo2, (float*)d_out, n);
}